// ODEINTWEvents_15547781611924
// MI455X (gfx1250) — compile-verified
//
#include <hip/hip_runtime.h>
#include <hip/hip_bf16.h>

typedef __attribute__((ext_vector_type(2))) float v2f;
typedef __attribute__((ext_vector_type(8))) float v8f;

constexpr int D = 64;         // state dim
constexpr int H = 128;        // hidden dim
constexpr int NSTEPS = 64;    // Euler steps
constexpr int WAVES = 8;      // waves per block (wave32)
constexpr int ROWS_PER_BLOCK = WAVES * 16;
constexpr int HP = H + 2;     // padded row stride (dwords) for h scratch: breaks 64-bank alignment, stays 8B-aligned
constexpr int DP = D + 2;     // padded row stride for f scratch

// tanh via hardware transcendentals: tanh(x) = 1 - 2/(exp(2x)+1)
// v_exp_f32 is base-2, so scale by 2*log2(e). Saturates correctly to +/-1.
__device__ __forceinline__ float fast_tanh(float x) {
    const float e = __builtin_amdgcn_exp2f(x * 2.8853900817779268f);
    return 1.0f - 2.0f * __builtin_amdgcn_rcpf(e + 1.0f);
}

__launch_bounds__(256)
__global__ void ode_euler_wmma(const float* __restrict__ init_state,
                               const float* __restrict__ first_t,
                               const float* __restrict__ last_t,
                               const float* __restrict__ step_size,
                               const float* __restrict__ W1,
                               const float* __restrict__ b1,
                               const float* __restrict__ W2,
                               const float* __restrict__ b2,
                               float* __restrict__ out,
                               int B) {
    // Weights stored PAIR-SWIZZLED: sW[((k>>1)*N + n)*2 + (k&1)] = W[k][n]
    // so a WMMA B-fragment {W[k0][n], W[k0+1][n]} is one contiguous 8B ds_load_b64.
    __shared__ float sW1[D * H];            // 32 KB
    __shared__ float sW2[H * D];            // 32 KB
    __shared__ float sH[WAVES][16 * HP];    // 65 KB, per-wave hidden/f scratch (padded rows)

    const int tid = threadIdx.x;

    // ---- stage weights to LDS, swizzling into pair layout (once) ----
    for (int i = tid; i < D * H; i += 256) {         // W1: [D][H], k = i/H, n = i%H
        const int k = i >> 7;
        const int n = i & (H - 1);
        sW1[((((k >> 1) * H) + n) << 1) | (k & 1)] = W1[i];
    }
    for (int i = tid; i < H * D; i += 256) {         // W2: [H][D], k = i/D, n = i%D
        const int k = i >> 6;
        const int n = i & (D - 1);
        sW2[((((k >> 1) * D) + n) << 1) | (k & 1)] = W2[i];
    }
    __syncthreads();

    const int wave = tid >> 5;
    const int lane = tid & 31;
    const int lh   = lane & 15;   // M / N index within 16
    const int hf   = lane >> 4;   // half-wave selector
    float* hbuf = sH[wave];

    const int rowA = blockIdx.x * ROWS_PER_BLOCK + wave * 16 + lh;

    // per-lane bias copies (N = n*16 + lh)
    float bias1[8], bias2[4];
#pragma unroll
    for (int n = 0; n < 8; ++n) bias1[n] = b1[n * 16 + lh];
#pragma unroll
    for (int n = 0; n < 4; ++n) bias2[n] = b2[n * 16 + lh];

    // ---- load state y as WMMA A-fragments (16 chunks of K=4) ----
    // A layout: lane holds M=lh, K = kc*4 + 2*hf + {0,1}
    v2f yA[16];
#pragma unroll
    for (int kc = 0; kc < 16; ++kc)
        yA[kc] = *(const v2f*)&init_state[(size_t)rowA * D + kc * 4 + 2 * hf];

    float t = first_t[rowA];
    const float lastv = last_t[rowA];
    const float dtv   = step_size[rowA];

    const size_t suBase    = (size_t)B * D;
    const size_t nssBase   = suBase + (size_t)B * NSTEPS;
    const size_t stepsBase = nssBase + (size_t)B;

    for (int s = 0; s < NSTEPS; ++s) {
        const float m = (t + 1e-5f < lastv) ? 1.0f : 0.0f;
        const float g = m * dtv;

        // ---------- GEMM1: h = tanh(y @ W1 + b1), h: [16, H] ----------
#pragma unroll
        for (int n = 0; n < 8; ++n) {
            v8f acc;
#pragma unroll
            for (int j = 0; j < 8; ++j) acc[j] = bias1[n];
#pragma unroll
            for (int kc = 0; kc < 16; ++kc) {
                // B frag: K = kc*4 + 2*hf + {0,1}, N = n*16 + lh  -> contiguous b64 in swizzled layout
                const v2f bb = *(const v2f*)&sW1[(((kc * 2 + hf) * H) + n * 16 + lh) * 2];
                acc = __builtin_amdgcn_wmma_f32_16x16x4_f32(
                        false, yA[kc], false, bb, (short)0, acc, false, false);
            }
            // C layout -> row-major (padded) h[16][HP] in LDS, with fast tanh
#pragma unroll
            for (int j = 0; j < 8; ++j)
                hbuf[(j + 8 * hf) * HP + n * 16 + lh] = fast_tanh(acc[j]);
        }

        // ---------- GEMM2: f = h @ W2 + b2, f: [16, D] ----------
        v8f acc2[4];
#pragma unroll
        for (int n = 0; n < 4; ++n)
#pragma unroll
            for (int j = 0; j < 8; ++j) acc2[n][j] = bias2[n];

#pragma unroll
        for (int kc = 0; kc < 32; ++kc) {
            const int k0 = kc * 4 + 2 * hf;
            // A frag of h from LDS: M=lh, K=k0+{0,1} (8B aligned, padded stride)
            const v2f aa = *(const v2f*)&hbuf[lh * HP + k0];
#pragma unroll
            for (int n = 0; n < 4; ++n) {
                const v2f bb = *(const v2f*)&sW2[(((kc * 2 + hf) * D) + n * 16 + lh) * 2];
                acc2[n] = __builtin_amdgcn_wmma_f32_16x16x4_f32(
                            false, aa, false, bb, (short)0, acc2[n], false, false);
            }
        }

        // stash f (C layout -> row-major [16][DP]) into per-wave LDS scratch
#pragma unroll
        for (int n = 0; n < 4; ++n)
#pragma unroll
            for (int j = 0; j < 8; ++j)
                hbuf[(j + 8 * hf) * DP + n * 16 + lh] = acc2[n][j];

        // y += (mask*dt) * f, reading f back in A-fragment layout
#pragma unroll
        for (int kc = 0; kc < 16; ++kc) {
            const v2f fv = *(const v2f*)&hbuf[lh * DP + kc * 4 + 2 * hf];
            yA[kc].x += g * fv.x;
            yA[kc].y += g * fv.y;
        }

        // per-step auxiliary outputs (divergence only after all WMMAs this step)
        if (hf == 0) out[suBase    + (size_t)rowA * NSTEPS + s] = 0.0f;
        else         out[stepsBase + (size_t)rowA * NSTEPS + s] = m;

        t += dtv;
    }

    // ---- final state (A-fragment layout -> row-major global) ----
#pragma unroll
    for (int kc = 0; kc < 16; ++kc) {
        float2 v; v.x = yA[kc].x; v.y = yA[kc].y;
        *(float2*)&out[(size_t)rowA * D + kc * 4 + 2 * hf] = v;
    }
    if (hf == 0) out[nssBase + rowA] = dtv;
}

extern "C" void kernel_launch(void* const* d_in, const int* in_sizes, int n_in,
                              void* d_out, int out_size, void* d_ws, size_t ws_size,
                              hipStream_t stream) {
    const float* init_state = (const float*)d_in[0];
    const float* first_t    = (const float*)d_in[1];
    const float* last_t     = (const float*)d_in[2];
    const float* step_size  = (const float*)d_in[3];
    const float* W1         = (const float*)d_in[4];
    const float* b1         = (const float*)d_in[5];
    const float* W2         = (const float*)d_in[6];
    const float* b2         = (const float*)d_in[7];

    const int B = in_sizes[0] / D;   // 32768
    dim3 grid(B / ROWS_PER_BLOCK);   // 256 blocks
    dim3 block(256);                 // 8 waves (wave32)
    ode_euler_wmma<<<grid, block, 0, stream>>>(init_state, first_t, last_t, step_size,
                                               W1, b1, W2, b2, (float*)d_out, B);
}